// RandomShiftsAugNew_438086664704
// MI455X (gfx1250) — compile-verified
//
#include <hip/hip_runtime.h>
#include <hip/hip_bf16.h>
#include <math.h>
#include <stdint.h>

#define PADV 4
#define HH   224
#define WW   224
#define CC   9
#define H2   (HH + 2 * PADV)   // 232

// ---- CDNA5 async global->LDS path (probe-confirmed present on this toolchain) ----
#if defined(__has_builtin)
#if __has_builtin(__builtin_amdgcn_global_load_async_to_lds_b128)
#define RSA_HAVE_ASYNC_LDS 1
#endif
#endif
#ifndef RSA_HAVE_ASYNC_LDS
#define RSA_HAVE_ASYNC_LDS 0
#endif

typedef int v4i __attribute__((vector_size(16)));

__global__ __launch_bounds__(256) void
RandomShiftsAugNew_438086664704_kernel(const float* __restrict__ x,
                                       const int* __restrict__ shift,
                                       float* __restrict__ out)
{
    // 18 rows (9 channels x {y0,y1}) of 224 floats each = 15.75 KB LDS
    __shared__ __align__(16) float smem[2 * CC * WW];

    const int tid = threadIdx.x;
    const int blk = blockIdx.x;
    const int n = blk / HH;          // batch index
    const int i = blk - n * HH;      // output row

    const int sx = shift[2 * n + 0]; // grid[...,0] -> width shift
    const int sy = shift[2 * n + 1]; // grid[...,1] -> height shift

    // Reference coordinate math, fp32 (H2 == W2 == 232); uniform -> SALU float pipe
    const float eps   = 1.0f / (float)H2;
    const float delta = (2.0f - 2.0f * eps) / (float)(H2 - 1); // linspace step
    const float scale = eps + eps;                              // 1/H2 + 1/W2

    const float ah  = -1.0f + eps + (float)i * delta;
    const float gy  = ah + (float)sy * scale;
    const float iy  = ((gy + 1.0f) * (float)H2 - 1.0f) * 0.5f;
    const float y0f = floorf(iy);
    const float y1f = y0f + 1.0f;
    const float wy1 = iy - y0f;
    const float wy0 = 1.0f - wy1;
    const bool  vy0 = (y0f >= 0.0f) && (y0f <= (float)(H2 - 1));
    const bool  vy1 = (y1f >= 0.0f) && (y1f <= (float)(H2 - 1));
    // padded-row index -> source-row index (replicate pad == clamp)
    const int yc0 = min(max((int)y0f, 0), H2 - 1);
    const int yc1 = min(max((int)y1f, 0), H2 - 1);
    const int rm0 = min(max(yc0 - PADV, 0), HH - 1);
    const int rm1 = min(max(yc1 - PADV, 0), HH - 1);

    const int base_nc = n * CC;

    // ---- Stage 18 source rows into LDS: 18 rows * 56 float4 = 1008 B128 transfers.
    // Up to 4 async transfers per thread (1008 = 3*256 + 240); only the 4th guarded.
    const int nvec = 2 * CC * (WW / 4); // 1008
#pragma unroll
    for (int k = 0; k < 4; ++k) {
        const int idx = tid + k * 256;
        if (k < 3 || idx < nvec) {
            const int r  = idx / (WW / 4);      // 0..17  (channel*2 + corner)
            const int v  = idx - r * (WW / 4);  // 0..55  (float4 within row)
            const int c  = r >> 1;
            const int rm = (r & 1) ? rm1 : rm0;
            const float* gp = x + (((base_nc + c) * HH) + rm) * WW + v * 4;
            float*       lp = smem + r * WW + v * 4;
#if RSA_HAVE_ASYNC_LDS
            __builtin_amdgcn_global_load_async_to_lds_b128(
                (__attribute__((address_space(1))) v4i*)(uintptr_t)gp,
                (__attribute__((address_space(3))) v4i*)lp,
                0, 0);
#else
            *(float4*)lp = *(const float4*)gp;
#endif
        }
    }

#if RSA_HAVE_ASYNC_LDS
#if defined(__has_builtin) && __has_builtin(__builtin_amdgcn_s_wait_asynccnt)
    __builtin_amdgcn_s_wait_asynccnt(0);
#else
    asm volatile("s_wait_asynccnt 0" ::: "memory");
#endif
#endif
    __syncthreads();

    // ---- Bilinear blend from LDS; one thread per output column ----
    if (tid < WW) {
        const int   j   = tid;
        const float aw  = -1.0f + eps + (float)j * delta;
        const float gx  = aw + (float)sx * scale;
        const float ix  = ((gx + 1.0f) * (float)H2 - 1.0f) * 0.5f;
        const float x0f = floorf(ix);
        const float x1f = x0f + 1.0f;
        const float wx1 = ix - x0f;
        const float wx0 = 1.0f - wx1;
        const bool  vx0 = (x0f >= 0.0f) && (x0f <= (float)(H2 - 1));
        const bool  vx1 = (x1f >= 0.0f) && (x1f <= (float)(H2 - 1));
        const int xc0 = min(max((int)x0f, 0), H2 - 1);
        const int xc1 = min(max((int)x1f, 0), H2 - 1);
        const int xm0 = min(max(xc0 - PADV, 0), WW - 1);
        const int xm1 = min(max(xc1 - PADV, 0), WW - 1);

        // zeros-padding validity folded into the 4 corner weights
        const float w00 = (vy0 && vx0) ? (wy0 * wx0) : 0.0f;
        const float w01 = (vy0 && vx1) ? (wy0 * wx1) : 0.0f;
        const float w10 = (vy1 && vx0) ? (wy1 * wx0) : 0.0f;
        const float w11 = (vy1 && vx1) ? (wy1 * wx1) : 0.0f;

#pragma unroll
        for (int c = 0; c < CC; ++c) {
            const float* r0 = smem + (2 * c + 0) * WW;
            const float* r1 = smem + (2 * c + 1) * WW;
            const float acc = w00 * r0[xm0] + w01 * r0[xm1]
                            + w10 * r1[xm0] + w11 * r1[xm1];
            out[(((base_nc + c) * HH) + i) * WW + j] = acc;
        }
    }
}

extern "C" void kernel_launch(void* const* d_in, const int* in_sizes, int n_in,
                              void* d_out, int out_size, void* d_ws, size_t ws_size,
                              hipStream_t stream) {
    const float* x     = (const float*)d_in[0];
    const int*   shift = (const int*)d_in[1];
    float*       out   = (float*)d_out;

    const int n = in_sizes[1] / 2;   // shift has 2 ints per batch element
    dim3 grid(n * HH);               // one block per (batch, output row)
    dim3 block(256);                 // 8 wave32s
    RandomShiftsAugNew_438086664704_kernel<<<grid, block, 0, stream>>>(x, shift, out);
}